// DySimODEFunc_18202071400766
// MI455X (gfx1250) — compile-verified
//
#include <hip/hip_runtime.h>
#include <hip/hip_bf16.h>
#include <math.h>

typedef __attribute__((ext_vector_type(2))) float v2f;
typedef __attribute__((ext_vector_type(8))) float v8f;

#define D_FEAT 64

// float atomic max via sign-split int/uint atomics (correct for IEEE ordering,
// arrays initialized to -inf bits).
__device__ __forceinline__ void atomic_max_f32(float* addr, float val) {
    if (val >= 0.0f) atomicMax((int*)addr, __float_as_int(val));
    else             atomicMin((unsigned int*)addr, (unsigned int)__float_as_int(val));
}

__global__ void init_ws_kernel(float* __restrict__ w, int n4) {
    int i = blockIdx.x * blockDim.x + threadIdx.x;
    if (i < n4) {
        if (i < (n4 >> 1)) ((int*)w)[i] = 0xFF800000; // -inf for the two max arrays
        else               w[i] = 0.0f;               // zeros for the two sum arrays
    }
}

__global__ void zero_out_kernel(float* __restrict__ out, int n) {
    int i = blockIdx.x * blockDim.x + threadIdx.x;
    if (i < n) out[i] = 0.0f;
}

__global__ void seg_max_kernel(const long long* __restrict__ frm,
                               const long long* __restrict__ to,
                               const float* __restrict__ attrs,
                               float* __restrict__ max_in,
                               float* __restrict__ max_out, int E) {
    int e = blockIdx.x * blockDim.x + threadIdx.x;
    if (e < E) {
        float a = attrs[e];
        atomic_max_f32(&max_in[(int)to[e]], a);
        atomic_max_f32(&max_out[(int)frm[e]], a);
    }
}

__global__ void exp_sum_kernel(const long long* __restrict__ frm,
                               const long long* __restrict__ to,
                               const float* __restrict__ attrs,
                               const float* __restrict__ max_in,
                               const float* __restrict__ max_out,
                               float* __restrict__ sum_in,
                               float* __restrict__ sum_out, int E) {
    int e = blockIdx.x * blockDim.x + threadIdx.x;
    if (e < E) {
        float a = attrs[e];
        int t = (int)to[e], f = (int)frm[e];
        atomicAdd(&sum_in[t],  __expf(a - max_in[t]));
        atomicAdd(&sum_out[f], __expf(a - max_out[f]));
    }
}

// One wave32 handles a tile of 16 edges x 64 features.
// msg(16x64) = diag(norm16) x Xgathered(16x64), computed as four 16-feature
// column tiles, each accumulated with four V_WMMA_F32_16X16X4_F32 (diagonal
// split into 16x4 K-chunks). Then scatter-add the D matrix with f32 atomics.
__global__ void __launch_bounds__(256)
scatter_wmma_kernel(const float* __restrict__ x,
                    const long long* __restrict__ frm_idx,
                    const long long* __restrict__ to_idx,
                    const float* __restrict__ attrs,
                    const float* __restrict__ max_in,
                    const float* __restrict__ max_out,
                    const float* __restrict__ sum_in,
                    const float* __restrict__ sum_out,
                    float* __restrict__ out, int E) {
    const int lane = threadIdx.x & 31;
    const int half = lane >> 4;   // 0: lanes 0-15, 1: lanes 16-31
    const int lm   = lane & 15;
    const int waves_per_block = blockDim.x >> 5;
    const int wave   = blockIdx.x * waves_per_block + (threadIdx.x >> 5);
    const int nwaves = gridDim.x * waves_per_block;
    const int ntiles = (E + 15) >> 4;

    for (int tile = wave; tile < ntiles; tile += nwaves) {
        int e = tile * 16 + lm;
        int frm = 0, to = 0;
        float norm = 0.0f;
        if (lane < 16 && e < E) {
            frm = (int)frm_idx[e];
            to  = (int)to_idx[e];
            float a = attrs[e];
            __builtin_prefetch(x + (size_t)frm * D_FEAT, 0, 3); // global_prefetch_b8
            float ein  = __expf(a - max_in[to])  / sum_in[to];
            float eout = __expf(a - max_out[frm]) / sum_out[frm];
            norm = sqrtf(ein * eout);
        }
        // lane l gets norm of tile-row (l&15); rows/targets broadcast on demand
        float norm_m = __shfl(norm, lm, 32);

        #pragma unroll
        for (int j = 0; j < 4; ++j) {          // 4 feature tiles of 16
            v8f acc = {0.f,0.f,0.f,0.f,0.f,0.f,0.f,0.f};
            #pragma unroll
            for (int c = 0; c < 4; ++c) {      // 4 K-chunks of the diagonal
                v2f A, B;
                #pragma unroll
                for (int v = 0; v < 2; ++v) {
                    // A (16x4 f32): VGPR v, lane l -> (m = l&15, k = v + 2*(l>>4))
                    int k = v + 2 * half;
                    A[v] = (lm == c * 4 + k) ? norm_m : 0.0f;
                    // B (4x16 f32): VGPR v, lane l -> (row k, col n = l&15)
                    int row = c * 4 + k;
                    int fr  = __shfl(frm, row, 32);
                    B[v] = x[(size_t)fr * D_FEAT + j * 16 + lm];
                }
                acc = __builtin_amdgcn_wmma_f32_16x16x4_f32(
                        false, A, false, B, (short)0, acc, false, false);
            }
            // D (16x16 f32): VGPR r, lane l -> (M = r + 8*(l>>4), N = l&15)
            #pragma unroll
            for (int r = 0; r < 8; ++r) {
                int M = r + 8 * half;
                int t = __shfl(to, M, 32);
                atomicAdd(&out[(size_t)t * D_FEAT + j * 16 + lm], acc[r]);
            }
        }
    }
}

extern "C" void kernel_launch(void* const* d_in, const int* in_sizes, int n_in,
                              void* d_out, int out_size, void* d_ws, size_t ws_size,
                              hipStream_t stream) {
    // inputs: [0]=t (unused), [1]=x f32 [N,64], [2]=edge_index i64 [2,E], [3]=edge_attrs f32 [E]
    const float*      x     = (const float*)d_in[1];
    const long long*  ei    = (const long long*)d_in[2];
    const float*      attrs = (const float*)d_in[3];
    const int E = in_sizes[3];
    const int N = out_size / D_FEAT;

    float* w       = (float*)d_ws;
    float* max_in  = w;
    float* max_out = w + (size_t)N;
    float* sum_in  = w + (size_t)2 * N;
    float* sum_out = w + (size_t)3 * N;
    float* out     = (float*)d_out;
    const long long* frm = ei;       // row 0
    const long long* to  = ei + E;   // row 1

    const int n4 = 4 * N;
    init_ws_kernel<<<(n4 + 255) / 256, 256, 0, stream>>>(w, n4);
    zero_out_kernel<<<(out_size + 255) / 256, 256, 0, stream>>>(out, out_size);
    seg_max_kernel<<<(E + 255) / 256, 256, 0, stream>>>(frm, to, attrs, max_in, max_out, E);
    exp_sum_kernel<<<(E + 255) / 256, 256, 0, stream>>>(frm, to, attrs, max_in, max_out,
                                                        sum_in, sum_out, E);
    const int ntiles  = (E + 15) / 16;
    const int nblocks = (ntiles + 7) / 8;   // 8 waves of 32 per 256-thread block
    scatter_wmma_kernel<<<nblocks, 256, 0, stream>>>(x, frm, to, attrs, max_in, max_out,
                                                     sum_in, sum_out, out, E);
}